// LungSsdBasic_46170898432412
// MI455X (gfx1250) — compile-verified
//
#include <hip/hip_runtime.h>
#include <stdint.h>

// SSD detection post-process for MI455X (gfx1250, wave32).
// Kernel 1: per-batch exact top-200 via 8-bit radix select (4 histogram
//           passes, L2-resident re-reads) + stable gather + bitonic sort.
// Kernel 2: decode 200 candidates, 208x208 pairwise area-sum via
//           V_WMMA_F32_16X16X4_F32 (A=[area_i,1], B=[1;area_j] rank-2 trick),
//           division-free IoU threshold -> LDS bitmask, sequential greedy NMS,
//           stable compaction, write (B,2,200,5).

#define TOPK 200
#define PADK 208          // 13 * 16
#define NTILE 13
#define MASKW 7           // ceil(208/32)
#define CONF_TH 0.01f
#define NMS_TH 0.45f
#define VAR0 0.1f
#define VAR1 0.2f

typedef __attribute__((ext_vector_type(2))) float v2f;
typedef __attribute__((ext_vector_type(8))) float v8f;

__device__ __forceinline__ unsigned scoreKey(float s) {
    // scores are >= 0 -> IEEE bits are monotone. Invalid (<= thresh) -> key 0,
    // which reproduces top_k's "-inf ties broken by lowest index" behavior.
    return (s > CONF_TH) ? __float_as_uint(s) : 0u;
}

// ---------------------------------------------------------------------------
// Kernel 1: one 1024-thread block per batch. Exact top-200 selection.
// ---------------------------------------------------------------------------
__global__ __launch_bounds__(1024) void topk_select(
    const float* __restrict__ conf, int P,
    int* __restrict__ topIdx, float* __restrict__ topScore)
{
    const int b   = blockIdx.x;
    const int tid = threadIdx.x;
    const int NT  = 1024;
    const float* sc = conf + (size_t)b * (size_t)P * 2;  // class-1 score at [p*2+1]

    __shared__ unsigned hist[256];
    __shared__ unsigned s_prefix;
    __shared__ int      s_krem;
    __shared__ unsigned s_cnt;
    __shared__ unsigned tcnt[1024];
    __shared__ unsigned long long skey[256];

    if (tid == 0) { s_prefix = 0u; s_krem = TOPK; }
    __syncthreads();

    // ---- 4-pass radix select: find exact 200th-largest key ----
    unsigned prefix = 0u, pmask = 0u;
    for (int pass = 0; pass < 4; ++pass) {
        const int shift = 24 - 8 * pass;
        for (int i = tid; i < 256; i += NT) hist[i] = 0u;
        __syncthreads();
        for (int p = tid; p < P; p += NT) {
            unsigned key = scoreKey(sc[p * 2 + 1]);
            if ((key & pmask) == prefix)
                atomicAdd(&hist[(key >> shift) & 255u], 1u);
        }
        __syncthreads();
        if (tid == 0) {
            int krem = s_krem;
            unsigned cum = 0; int bsel = 0;
            for (int bkt = 255; bkt >= 0; --bkt) {
                unsigned c = hist[bkt];
                if (cum + c >= (unsigned)krem || bkt == 0) {
                    bsel = bkt; s_krem = krem - (int)cum; break;
                }
                cum += c;
            }
            s_prefix = prefix | ((unsigned)bsel << shift);
        }
        __syncthreads();
        prefix = s_prefix;
        pmask |= (0xFFu << shift);
    }
    const unsigned T = prefix;      // threshold key
    const int kEq = s_krem;         // how many == T to take (lowest index first)

    // ---- gather ----
    for (int i = tid; i < 256; i += NT) skey[i] = 0ull;
    if (tid == 0) s_cnt = 0u;
    __syncthreads();

    // strictly greater: unordered slots (fixed up by the full sort below)
    for (int p = tid; p < P; p += NT) {
        unsigned key = scoreKey(sc[p * 2 + 1]);
        if (key > T) {
            unsigned pos = atomicAdd(&s_cnt, 1u);
            if (pos < TOPK)
                skey[pos] = ((unsigned long long)key << 32) |
                            (unsigned)(0xFFFFFFFFu - (unsigned)p);
        }
    }
    __syncthreads();
    int nGreater = (int)s_cnt;
    if (nGreater > TOPK) nGreater = TOPK;

    // equal-to-T, in ascending index order: contiguous chunks + exclusive scan
    const int CHUNK = (P + NT - 1) / NT;
    const int cbase = tid * CHUNK;
    unsigned myCnt = 0;
    for (int i = 0; i < CHUNK; ++i) {
        int p = cbase + i;
        if (p < P && scoreKey(sc[p * 2 + 1]) == T) myCnt++;
    }
    tcnt[tid] = myCnt;
    __syncthreads();
    if (tid == 0) {
        unsigned run = 0;
        for (int i = 0; i < NT; ++i) { unsigned c = tcnt[i]; tcnt[i] = run; run += c; }
    }
    __syncthreads();
    unsigned rank = tcnt[tid];
    for (int i = 0; i < CHUNK && rank < (unsigned)kEq; ++i) {
        int p = cbase + i;
        if (p < P && scoreKey(sc[p * 2 + 1]) == T) {
            int slot = nGreater + (int)rank;
            if (slot < TOPK)
                skey[slot] = ((unsigned long long)T << 32) |
                             (unsigned)(0xFFFFFFFFu - (unsigned)p);
            rank++;
        }
    }
    __syncthreads();

    // ---- bitonic sort, 256 elements, descending composite key ----
    for (int k = 2; k <= 256; k <<= 1) {
        for (int j = k >> 1; j > 0; j >>= 1) {
            __syncthreads();
            for (int idx = tid; idx < 256; idx += NT) {
                int q = idx ^ j;
                if (q > idx) {
                    unsigned long long a = skey[idx], c = skey[q];
                    bool desc = ((idx & k) == 0);
                    if ((a < c) == desc) { skey[idx] = c; skey[q] = a; }
                }
            }
        }
    }
    __syncthreads();

    if (tid < TOPK) {
        unsigned long long e = skey[tid];
        unsigned key = (unsigned)(e >> 32);
        unsigned pid = 0xFFFFFFFFu - (unsigned)e;
        if (pid >= (unsigned)P) pid = 0u;  // degenerate padding guard
        topIdx[b * TOPK + tid]   = (int)pid;
        topScore[b * TOPK + tid] = (key != 0u) ? __uint_as_float(key) : 0.0f;
    }
}

// ---------------------------------------------------------------------------
// Kernel 2: one 256-thread block (8 wave32) per batch. Decode + WMMA-assisted
// IoU bitmask + sequential greedy NMS + stable compaction.
// ---------------------------------------------------------------------------
__global__ __launch_bounds__(256) void nms_kernel(
    const float* __restrict__ loc, const float* __restrict__ priors, int P,
    const int* __restrict__ topIdx, const float* __restrict__ topScore,
    float* __restrict__ out)
{
    const int b   = blockIdx.x;
    const int tid = threadIdx.x;

    __shared__ float sx1[PADK], sy1[PADK], sx2[PADK], sy2[PADK];
    __shared__ float sarea[PADK], sscore[PADK];
    __shared__ unsigned smask[PADK * MASKW];   // iou>thresh bits, 7 words/row
    __shared__ int spos[TOPK];

    // decode the 200 candidates (padding rows 200..207 -> degenerate zeros)
    for (int t = tid; t < PADK; t += 256) {
        if (t < TOPK) {
            int p   = topIdx[b * TOPK + t];
            float s = topScore[b * TOPK + t];
            float4 l  = ((const float4*)loc)[(size_t)b * P + p];
            float4 pr = ((const float4*)priors)[p];
            float cx = pr.x + l.x * VAR0 * pr.z;
            float cy = pr.y + l.y * VAR0 * pr.w;
            float w  = pr.z * expf(l.z * VAR1);
            float h  = pr.w * expf(l.w * VAR1);
            float x1 = cx - 0.5f * w, y1 = cy - 0.5f * h;
            float x2 = x1 + w,        y2 = y1 + h;
            sx1[t] = x1; sy1[t] = y1; sx2[t] = x2; sy2[t] = y2;
            sarea[t]  = (x2 - x1) * (y2 - y1);
            sscore[t] = s;
        } else {
            sx1[t] = sy1[t] = sx2[t] = sy2[t] = 0.f;
            sarea[t] = 0.f; sscore[t] = -1.f;
        }
    }
    for (int i = tid; i < PADK * MASKW; i += 256) smask[i] = 0u;
    __syncthreads();

    // pairSum[i][j] = area_i + area_j via V_WMMA_F32_16X16X4_F32:
    //   A(16x4) = [area_i, 1, 0, 0] rows, B(4x16) = [1; area_j; 0; 0] cols.
    // ISA layouts (§7.12.2): A lanes0-15 hold K=0(v0)/K=1(v1), lanes16-31 K=2/3;
    // B lanes0-15 hold K=0(v0)/K=1(v1) for N=lane; C/D: M = r + 8*half, N = lane&15.
    const int lane = tid & 31;
    const int wv   = tid >> 5;
    const int half = lane >> 4;
    const int mn   = lane & 15;
    for (int tile = wv; tile < NTILE * NTILE; tile += 8) {
        const int ti = tile / NTILE, tj = tile % NTILE;
        v2f a, bm;
        a.x  = half ? 0.f : sarea[ti * 16 + mn];
        a.y  = half ? 0.f : 1.0f;
        bm.x = half ? 0.f : 1.0f;
        bm.y = half ? 0.f : sarea[tj * 16 + mn];
        v8f c = {};
        c = __builtin_amdgcn_wmma_f32_16x16x4_f32(
                /*neg_a=*/false, a, /*neg_b=*/false, bm,
                /*c_mod=*/(short)0, c, /*reuse_a=*/false, /*reuse_b=*/false);
        #pragma unroll
        for (int r = 0; r < 8; ++r) {
            const int i = ti * 16 + r + 8 * half;
            const int j = tj * 16 + mn;
            const float ps = c[r];  // area_i + area_j
            float ww = fminf(sx2[i], sx2[j]) - fmaxf(sx1[i], sx1[j]);
            float hh = fminf(sy2[i], sy2[j]) - fmaxf(sy1[i], sy1[j]);
            float inter = fmaxf(ww, 0.f) * fmaxf(hh, 0.f);
            // iou > TH  <=>  inter > TH * (pairSum - inter), union > 0
            bool bit = (i < TOPK) && (j < TOPK) && (inter > NMS_TH * (ps - inter));
            if (bit) atomicOr(&smask[i * MASKW + (j >> 5)], 1u << (j & 31));
        }
    }
    __syncthreads();

    // sequential greedy NMS over bitmask rows; stable output positions
    if (tid == 0) {
        unsigned sup[MASKW] = {0, 0, 0, 0, 0, 0, 0};
        int pos = 0;
        for (int t = 0; t < TOPK; ++t) {
            bool v = sscore[t] > CONF_TH;
            bool s = (sup[t >> 5] >> (t & 31)) & 1u;
            bool k = v && !s;
            spos[t] = k ? pos++ : -1;
            if (k) {
                #pragma unroll
                for (int w2 = 0; w2 < MASKW; ++w2) sup[w2] |= smask[t * MASKW + w2];
            }
        }
    }
    __syncthreads();

    // output: (B, 2, 200, 5); class 0 all zeros, class 1 = kept rows compacted
    float* ob = out + (size_t)b * 2 * TOPK * 5;
    for (int i = tid; i < 2 * TOPK * 5; i += 256) ob[i] = 0.f;
    __syncthreads();
    if (tid < TOPK && spos[tid] >= 0) {
        float* row = ob + (size_t)TOPK * 5 + (size_t)spos[tid] * 5;
        row[0] = sscore[tid];
        row[1] = sx1[tid]; row[2] = sy1[tid];
        row[3] = sx2[tid]; row[4] = sy2[tid];
    }
}

// ---------------------------------------------------------------------------
extern "C" void kernel_launch(void* const* d_in, const int* in_sizes, int n_in,
                              void* d_out, int out_size, void* d_ws, size_t ws_size,
                              hipStream_t stream)
{
    const float* loc    = (const float*)d_in[0];   // (B, P, 4)
    const float* conf   = (const float*)d_in[1];   // (B, P, 2)
    const float* priors = (const float*)d_in[2];   // (P, 4)
    const int P = in_sizes[2] / 4;
    const int B = in_sizes[0] / (P * 4);

    int*   topIdx   = (int*)d_ws;                                          // B*200 ints
    float* topScore = (float*)((char*)d_ws + (size_t)B * TOPK * sizeof(int)); // B*200 floats

    topk_select<<<B, 1024, 0, stream>>>(conf, P, topIdx, topScore);
    nms_kernel<<<B, 256, 0, stream>>>(loc, priors, P, topIdx, topScore, (float*)d_out);
}